// RORDefense_41369124995198
// MI455X (gfx1250) — compile-verified
//
#include <hip/hip_runtime.h>
#include <hip/hip_bf16.h>

typedef float v2f __attribute__((ext_vector_type(2)));
typedef float v8f __attribute__((ext_vector_type(8)));

#define KPTS   4096
#define NBATCH 16
#define R2     (1.1f * 1.1f)
#define NMIN   2

// One workgroup = 8 waves; wave w handles rows [i0, i0+16) of one batch.
// LDS holds the batch's point cloud as float4 {x, y, z, -0.5*||x||^2} (64 KB).
// Augmented GEMM: A row = [x,y,z,1], B col = [x,y,z,-0.5*sq]  =>
//   acc = dot(x_i,x_j) - 0.5*sq_j ;  keep pair iff acc >= 0.5*sq_i - 0.5*r^2.
__global__ __launch_bounds__(256) void ror_wmma_kernel(const float* __restrict__ x,
                                                       float* __restrict__ out,
                                                       float* __restrict__ mask) {
    __shared__ float4 xs[KPTS];   // 64 KB of the 320 KB/WGP LDS

    const int b    = blockIdx.x >> 5;      // 32 workgroups per batch
    const int tg   = blockIdx.x & 31;
    const int wave = threadIdx.x >> 5;
    const int lane = threadIdx.x & 31;
    const int half = lane >> 4;            // 0: K=0,1 ; 1: K=2,3 (and M-rows +8 in C/D)
    const int i0   = tg * 128 + wave * 16; // row-tile base (within batch)

    // ---- stage batch point cloud into LDS with w = -0.5*||p||^2 ----
    const float* xb = x + (size_t)b * KPTS * 3;
    for (int p = threadIdx.x; p < KPTS; p += 256) {
        float px = xb[3 * p + 0];
        float py = xb[3 * p + 1];
        float pz = xb[3 * p + 2];
        float w  = -0.5f * (px * px + py * py + pz * pz);
        xs[p] = make_float4(px, py, pz, w);
    }
    __syncthreads();

    const float2* xs2 = (const float2*)xs;
    const float*  xsf = (const float*)xs;

    // ---- A-tile (16x4 f32): lane holds point i0+(lane&15);
    //      low half supplies {x,y} (K0,K1), high half {z,1} (K2,K3=1).
    const int pi = i0 + (lane & 15);
    float2 af = xs2[pi * 2 + half];            // ds_load_b64
    v2f av;
    av.x = af.x;
    av.y = half ? 1.0f : af.y;                 // K3 := 1 for the augmented column

    // Loop-invariant per-row thresholds: ht = 0.5*sq_i - 0.5*r^2 = -w_i - 0.5*r^2.
    // Row M = v + 8*half; broadcast read of xs[i0+M].w (uniform per half).
    float ht[8];
#pragma unroll
    for (int v = 0; v < 8; ++v)
        ht[v] = -xsf[(i0 + half * 8 + v) * 4 + 3] - 0.5f * R2;

    int cnt[8] = {0, 0, 0, 0, 0, 0, 0, 0};

    // ---- sweep all 256 column tiles: ds_load_b64 -> wmma -> 8 cmp/count ----
    for (int jt = 0; jt < KPTS / 16; ++jt) {
        const int pj = jt * 16 + (lane & 15);
        float2 bf2 = xs2[pj * 2 + half];       // {x,y} or {z, -0.5*sq}
        v2f bv;  bv.x = bf2.x;  bv.y = bf2.y;

        v8f acc = {};
        acc = __builtin_amdgcn_wmma_f32_16x16x4_f32(
            /*neg_a=*/false, av, /*neg_b=*/false, bv,
            /*c_mod=*/(short)0, acc, /*reuse_a=*/false, /*reuse_b=*/false);

#pragma unroll
        for (int v = 0; v < 8; ++v)
            cnt[v] += (acc[v] >= ht[v]) ? 1 : 0;   // d2 <= r^2
    }

    // ---- reduce counts across each 16-lane N-group ----
#pragma unroll
    for (int v = 0; v < 8; ++v) {
#pragma unroll
        for (int off = 1; off < 16; off <<= 1)
            cnt[v] += __shfl_xor(cnt[v], off, 32);
    }

    // lanes 0..7 write rows 0..7, lanes 16..23 write rows 8..15
    const int idx = lane & 7;
    int c = cnt[0];
#pragma unroll
    for (int v = 1; v < 8; ++v) c = (idx == v) ? cnt[v] : c;

    if ((lane & 15) < 8) {
        const int row = half * 8 + idx;
        const int p   = i0 + row;
        const float s = (c >= NMIN) ? 1.0f : 0.0f;   // count includes self
        float4 pt = xs[p];
        const size_t g = (size_t)b * KPTS + p;
        out[3 * g + 0] = pt.x * s;
        out[3 * g + 1] = pt.y * s;
        out[3 * g + 2] = pt.z * s;
        mask[g] = s;
    }
}

extern "C" void kernel_launch(void* const* d_in, const int* in_sizes, int n_in,
                              void* d_out, int out_size, void* d_ws, size_t ws_size,
                              hipStream_t stream) {
    (void)in_sizes; (void)n_in; (void)d_ws; (void)ws_size; (void)out_size;
    const float* x = (const float*)d_in[0];
    float* out  = (float*)d_out;                         // [16,4096,3] zeroed points
    float* mask = out + (size_t)NBATCH * KPTS * 3;       // [16,4096] keep-mask as float
    ror_wmma_kernel<<<dim3(NBATCH * 32), dim3(256), 0, stream>>>(x, out, mask);
}